// GraphItem2Vec_36636071034881
// MI455X (gfx1250) — compile-verified
//
#include <hip/hip_runtime.h>
#include <hip/hip_bf16.h>

typedef __attribute__((ext_vector_type(2))) float v2f;
typedef __attribute__((ext_vector_type(8))) float v8f;

#define DIM 128
#define WT_STRIDE 132   // padded LDS stride: bank = (4n + k) % 64 -> conflict-free b64 reads

// ---------------- degree / normalization ----------------

__global__ void init_deg_kernel(float* deg, int V) {
    int v = blockIdx.x * blockDim.x + threadIdx.x;
    if (v < V) deg[v] = 1.0f;  // self-loop contribution
}

__global__ void edge_deg_kernel(const int* __restrict__ dst, float* deg, int E) {
    int e = blockIdx.x * blockDim.x + threadIdx.x;
    if (e < E) atomicAdd(&deg[dst[e]], 1.0f);
}

__global__ void dinv_kernel(float* deg, int V) {
    int v = blockIdx.x * blockDim.x + threadIdx.x;
    if (v < V) deg[v] = rsqrtf(deg[v]);  // deg >= 1 always (self loops)
}

// ---------------- H = X @ W via fp32 WMMA (16x16x4) ----------------
// One wave computes a 16-row x 128-col tile of H. W held transposed in LDS.

__global__ __launch_bounds__(256) void gemm_wmma_kernel(const float* __restrict__ X,
                                                        const float* __restrict__ W,
                                                        float* __restrict__ H, int V) {
    __shared__ float Wt[DIM * WT_STRIDE];
    for (int i = threadIdx.x; i < DIM * DIM; i += blockDim.x) {
        int k = i >> 7, n = i & (DIM - 1);
        Wt[n * WT_STRIDE + k] = W[i];
    }
    __syncthreads();

    const int wave = threadIdx.x >> 5;
    const int lane = threadIdx.x & 31;
    const int rowBase = (blockIdx.x * 8 + wave) * 16;
    if (rowBase >= V) return;

    const int loCol = lane & 15;
    const int khi   = (lane >> 4) * 2;  // ISA layout: lanes 0-15 -> K,K+1 ; lanes 16-31 -> K+2,K+3

    if (rowBase + 16 <= V) {
        v8f acc[8] = {};
        const float* Arow = X + (size_t)(rowBase + loCol) * DIM;
        for (int k = 0; k < DIM; k += 4) {
            v2f a;
            a.x = Arow[k + khi];
            a.y = Arow[k + khi + 1];
#pragma unroll
            for (int nt = 0; nt < 8; nt++) {
                const float* wp = &Wt[(nt * 16 + loCol) * WT_STRIDE + k + khi];
                v2f b;
                b.x = wp[0];
                b.y = wp[1];
                acc[nt] = __builtin_amdgcn_wmma_f32_16x16x4_f32(
                    false, a, false, b, (short)0, acc[nt], false, false);
            }
        }
        // C/D layout: VGPR r holds row (r + 8*(lane>=16)), col = lane&15
        const int rOff = (lane >> 4) * 8;
#pragma unroll
        for (int nt = 0; nt < 8; nt++) {
#pragma unroll
            for (int r = 0; r < 8; r++) {
                H[(size_t)(rowBase + rOff + r) * DIM + nt * 16 + loCol] = acc[nt][r];
            }
        }
    } else {
        // scalar tail for a partial 16-row tile (V not multiple of 16)
        for (int row = rowBase + lane; row < V; row += 32) {
            const float* Arow = X + (size_t)row * DIM;
            for (int n = 0; n < DIM; n++) {
                float s = 0.0f;
                for (int k = 0; k < DIM; k++) s += Arow[k] * Wt[n * WT_STRIDE + k];
                H[(size_t)row * DIM + n] = s;
            }
        }
    }
}

// ---------------- nodes = dinv^2 * h + bias (self-loop term + bias) ----------------

__global__ void init_nodes_kernel(const float* __restrict__ h, const float* __restrict__ dinv,
                                  const float* __restrict__ bias, float* __restrict__ nodes,
                                  int V) {
    int i = blockIdx.x * blockDim.x + threadIdx.x;  // over V * 32 float4-granules
    if (i >= V * 32) return;
    int v = i >> 5;
    int c = (i & 31) * 4;
    float s = dinv[v];
    s = s * s;
    float4 hv = *reinterpret_cast<const float4*>(h + (size_t)v * DIM + c);
    float4 bv = *reinterpret_cast<const float4*>(bias + c);
    float4 o;
    o.x = s * hv.x + bv.x;
    o.y = s * hv.y + bv.y;
    o.z = s * hv.z + bv.z;
    o.w = s * hv.w + bv.w;
    *reinterpret_cast<float4*>(nodes + (size_t)v * DIM + c) = o;
}

// ---------------- edge scatter: nodes[dst] += dinv[src]*dinv[dst] * h[src] ----------------
// One wave per edge; each lane handles 4 contiguous floats (float4 gather, 4 f32 atomics).

__global__ __launch_bounds__(256) void scatter_kernel(const int* __restrict__ src,
                                                      const int* __restrict__ dst,
                                                      const float* __restrict__ h,
                                                      const float* __restrict__ dinv,
                                                      float* __restrict__ nodes, int E) {
    const int lane = threadIdx.x & 31;
    const int warp = (blockIdx.x * blockDim.x + threadIdx.x) >> 5;
    const int nwarps = (gridDim.x * blockDim.x) >> 5;
    for (int e = warp; e < E; e += nwarps) {
        int s = src[e];
        int d = dst[e];
        float w = dinv[s] * dinv[d];
        float4 hv = *reinterpret_cast<const float4*>(h + (size_t)s * DIM + lane * 4);
        float* np = nodes + (size_t)d * DIM + lane * 4;
        atomicAdd(np + 0, w * hv.x);
        atomicAdd(np + 1, w * hv.y);
        atomicAdd(np + 2, w * hv.z);
        atomicAdd(np + 3, w * hv.w);
    }
}

// ---------------- scores[b][s] = dot(nodes[items[b]], nodes[samples[b][s]]) ----------------
// One wave per batch item b; item vector cached in registers; wave32 xor-shuffle reduction.

__global__ __launch_bounds__(256) void score_kernel(const int* __restrict__ items,
                                                    const int* __restrict__ samples,
                                                    const float* __restrict__ nodes,
                                                    float* __restrict__ scores, int B, int S) {
    const int lane = threadIdx.x & 31;
    const int b = (blockIdx.x * blockDim.x + threadIdx.x) >> 5;
    if (b >= B) return;
    const int it = items[b];
    float4 iv = *reinterpret_cast<const float4*>(nodes + (size_t)it * DIM + lane * 4);
    const int* srow = samples + (size_t)b * S;
    for (int s = 0; s < S; s++) {
        int sm = srow[s];
        float4 sv = *reinterpret_cast<const float4*>(nodes + (size_t)sm * DIM + lane * 4);
        float p = iv.x * sv.x + iv.y * sv.y + iv.z * sv.z + iv.w * sv.w;
#pragma unroll
        for (int off = 16; off > 0; off >>= 1) p += __shfl_xor(p, off, 32);
        if (lane == 0) scores[(size_t)b * S + s] = p;
    }
}

// ---------------- launch ----------------

extern "C" void kernel_launch(void* const* d_in, const int* in_sizes, int n_in,
                              void* d_out, int out_size, void* d_ws, size_t ws_size,
                              hipStream_t stream) {
    const int*   items   = (const int*)d_in[0];
    const int*   samples = (const int*)d_in[1];
    const int*   edges   = (const int*)d_in[2];   // [2, E]: row 0 = src, row 1 = dst
    const float* emb     = (const float*)d_in[3];
    const float* W       = (const float*)d_in[4];
    const float* bias    = (const float*)d_in[5];
    float* scores = (float*)d_out;

    const int B = in_sizes[0];
    const int S = in_sizes[1] / B;
    const int E = in_sizes[2] / 2;
    const int V = in_sizes[3] / DIM;

    const int* edge_src = edges;
    const int* edge_dst = edges + E;

    // workspace: dinv [Vpad] | h [V*DIM] | nodes [V*DIM]
    float* dinv = (float*)d_ws;
    size_t Vpad = ((size_t)V + 255) & ~(size_t)255;
    float* h     = dinv + Vpad;
    float* nodes = h + (size_t)V * DIM;

    const int T = 256;

    // 1) degree (self-loops -> init 1) and edge counts, then rsqrt
    init_deg_kernel<<<(V + T - 1) / T, T, 0, stream>>>(dinv, V);
    edge_deg_kernel<<<(E + T - 1) / T, T, 0, stream>>>(edge_dst, dinv, E);
    dinv_kernel<<<(V + T - 1) / T, T, 0, stream>>>(dinv, V);

    // 2) h = emb @ W  (fp32 WMMA)
    {
        int rowTiles = (V + 15) / 16;
        int blocks = (rowTiles + 7) / 8;  // 8 waves/block, 1 tile/wave
        gemm_wmma_kernel<<<blocks, T, 0, stream>>>(emb, W, h, V);
    }

    // 3) nodes = dinv^2 * h + bias
    {
        long long tot = (long long)V * 32;
        init_nodes_kernel<<<(unsigned)((tot + T - 1) / T), T, 0, stream>>>(h, dinv, bias, nodes, V);
    }

    // 4) edge aggregation (wave per edge)
    {
        int blocks = (E + 7) / 8;  // 8 waves per block, grid-stride covers remainder
        if (blocks > 131072) blocks = 131072;
        scatter_kernel<<<blocks, T, 0, stream>>>(edge_src, edge_dst, h, dinv, nodes, E);
    }

    // 5) sampled scoring (wave per batch item)
    {
        int blocks = (B + 7) / 8;
        score_kernel<<<blocks, T, 0, stream>>>(items, samples, nodes, scores, B, S);
    }
}